// BeamSearch_41257455845859
// MI455X (gfx1250) — compile-verified
//
#include <hip/hip_runtime.h>

// Beam search (B=128, T=256, V=1024, K=3) for MI455X / gfx1250.
// One workgroup per sequence (the scan over T is serial), so the design
// goal is minimal per-step critical-path latency:
//   * depth-2 async global->LDS row prefetch (CDNA5 ASYNCcnt engine)
//   * 128 threads = 4 waves = 1 wave/SIMD32 -> full issue rate per wave
//   * packed u64 keys (value-desc, index-asc) -> 1 cmp per merge compare
//   * exact beam pruning: top-3 of score_k + logp_v  ==  {scores} x {top-3 e}
//   * parity-double-buffered reduction scratch -> only 2 barriers per step

#define T_   256
#define V_   1024
#define K_   3
#define NT_  128          // threads per block (4 waves, 1 per SIMD32)
#define NW_  (NT_ / 32)   // waves per block
#define VPT_ (V_ / NT_)   // elements per thread = 8
#define PF_  2            // prefetch depth (rows in flight)
#define NBUF_ 4           // LDS row ring buffers
#define EPS_F 2.220446049250313e-16f

static_assert(V_ == NT_ * VPT_, "row must tile exactly");

typedef unsigned long long u64;

// ---- packed sort key: value-descending, index-ascending -------------------
// e > 0 always here, so the IEEE bit pattern of e is order-isomorphic to e.
__device__ __forceinline__ u64 mk_key(float e, int v) {
  return ((u64)__float_as_uint(e) << 32) | (unsigned)((V_ - 1) - v);
}

__device__ __forceinline__ void insert3u(u64 k, u64& t0, u64& t1, u64& t2) {
  if (k > t2) {
    if (k > t0)      { t2 = t1; t1 = t0; t0 = k; }
    else if (k > t1) { t2 = t1; t1 = k; }
    else             { t2 = k; }
  }
}

// merge two sorted-descending triples -> top-3 (keys are unique)
__device__ __forceinline__ void merge3u(u64& a0, u64& a1, u64& a2,
                                        u64 b0, u64 b1, u64 b2) {
  u64 r0, r1, r2;
  if (a0 >= b0) {
    r0 = a0;
    if (a1 >= b0) { r1 = a1; r2 = (a2 >= b0) ? a2 : b0; }
    else          { r1 = b0; r2 = (a1 >= b1) ? a1 : b1; }
  } else {
    r0 = b0;
    if (b1 >= a0) { r1 = b1; r2 = (b2 >= a0) ? b2 : a0; }
    else          { r1 = a0; r2 = (a1 >= b1) ? a1 : b1; }
  }
  a0 = r0; a1 = r1; a2 = r2;
}

// ---- float (value desc, index asc) top-3 for the final 9-candidate pick ---
__device__ __forceinline__ bool fbetter(float av, int ai, float bv, int bi) {
  return (av > bv) || ((av == bv) && (ai < bi));
}
__device__ __forceinline__ void finsert3(float v, int i,
                                         float& v0, int& i0,
                                         float& v1, int& i1,
                                         float& v2, int& i2) {
  if (!fbetter(v, i, v2, i2)) return;
  if (fbetter(v, i, v0, i0)) { v2 = v1; i2 = i1; v1 = v0; i1 = i0; v0 = v; i0 = i; }
  else if (fbetter(v, i, v1, i1)) { v2 = v1; i2 = i1; v1 = v; i1 = i; }
  else { v2 = v; i2 = i; }
}

// ---- wave32 reductions ----------------------------------------------------
__device__ __forceinline__ float wred_max(float x) {
#pragma unroll
  for (int o = 16; o > 0; o >>= 1) x = fmaxf(x, __shfl_xor(x, o, 32));
  return x;
}
__device__ __forceinline__ float wred_sum(float x) {
#pragma unroll
  for (int o = 16; o > 0; o >>= 1) x += __shfl_xor(x, o, 32);
  return x;
}

__global__ __launch_bounds__(NT_) void beam_kernel(const float* __restrict__ logits,
                                                   int* __restrict__ tok_out,
                                                   float* __restrict__ sc_out) {
  __shared__ __align__(16) float rowbuf[NBUF_][V_];  // async-copy ring (16 KB)
  // reduction scratch, double-buffered by step parity -> no trailing barrier
  __shared__ float red_m[2][NW_];
  __shared__ float red_s[2][NW_];
  __shared__ u64   wkey[2][NW_][3];
  __shared__ short toks[T_ * K_];
  __shared__ unsigned char pars[T_ * K_];
  __shared__ float sc_fin[K_];

  const int tid  = threadIdx.x;
  const int wave = tid >> 5;
  const int lane = tid & 31;
  const int b    = blockIdx.x;
  const float* __restrict__ base = logits + (size_t)b * (T_ * V_);
  const int col = tid * VPT_;  // this thread's 8-element slice of each row

  // Beam scores live in thread 0's registers (init: [0, -inf, -inf])
  float s0 = 0.0f, s1 = -__builtin_inff(), s2 = -__builtin_inff();

  // ---- prologue: async-prefetch rows 0..PF_-1 into LDS (own slice only) ---
#pragma unroll
  for (int r = 0; r < PF_; ++r) {
    const float* g = base + r * V_ + col;
    unsigned l = (unsigned)(size_t)(const void*)&rowbuf[r][col];
    asm volatile("global_load_async_to_lds_b128 %0, %1, off\n\t"
                 "global_load_async_to_lds_b128 %0, %1, off offset:16"
                 :: "v"(l), "v"(g) : "memory");
  }

  for (int t = 0; t < T_; ++t) {
    const int p = t & 1;

    // Async loads complete in order: leaving <=2 outstanding guarantees
    // row t's two b128 transfers (each thread's own slice) are resident.
    if (t < T_ - 1) asm volatile("s_wait_asynccnt 2" ::: "memory");
    else            asm volatile("s_wait_asynccnt 0" ::: "memory");

    float x[VPT_];
    {
      const float4 a = *(const float4*)&rowbuf[t & (NBUF_ - 1)][col];
      const float4 c = *(const float4*)&rowbuf[t & (NBUF_ - 1)][col + 4];
      x[0] = a.x; x[1] = a.y; x[2] = a.z; x[3] = a.w;
      x[4] = c.x; x[5] = c.y; x[6] = c.z; x[7] = c.w;
    }

    // kick off the async copy of row t+PF_ into the ring
    if (t + PF_ < T_) {
      const float* g = base + (size_t)(t + PF_) * V_ + col;
      unsigned l = (unsigned)(size_t)(const void*)&rowbuf[(t + PF_) & (NBUF_ - 1)][col];
      asm volatile("global_load_async_to_lds_b128 %0, %1, off\n\t"
                   "global_load_async_to_lds_b128 %0, %1, off offset:16"
                   :: "v"(l), "v"(g) : "memory");
    }

    // ---- stage 1: block max --------------------------------------------
    float m = x[0];
#pragma unroll
    for (int j = 1; j < VPT_; ++j) m = fmaxf(m, x[j]);
    m = wred_max(m);
    if (lane == 0) red_m[p][wave] = m;
    __syncthreads();  // barrier 1: max -> exp

    // ---- stage 2: e = exp(x-max), sum, and block top-3 of e -------------
    // (log is strictly increasing, so top-3 of e == top-3 of log-prob,
    //  with the same index tie-break)
    float bm = red_m[p][0];
#pragma unroll
    for (int w = 1; w < NW_; ++w) bm = fmaxf(bm, red_m[p][w]);

    float e[VPT_];
    float s = 0.f;
#pragma unroll
    for (int j = 0; j < VPT_; ++j) { e[j] = __expf(x[j] - bm); s += e[j]; }
    s = wred_sum(s);
    if (lane == 0) red_s[p][wave] = s;

    u64 t0 = 0, t1 = 0, t2 = 0;  // 0 == "-inf" sentinel (real keys > 0)
#pragma unroll
    for (int j = 0; j < VPT_; ++j) insert3u(mk_key(e[j], col + j), t0, t1, t2);
#pragma unroll
    for (int o = 16; o > 0; o >>= 1) {
      u64 b0 = __shfl_xor(t0, o, 32);
      u64 b1 = __shfl_xor(t1, o, 32);
      u64 b2 = __shfl_xor(t2, o, 32);
      merge3u(t0, t1, t2, b0, b1, b2);
    }
    if (lane == 0) { wkey[p][wave][0] = t0; wkey[p][wave][1] = t1; wkey[p][wave][2] = t2; }
    __syncthreads();  // barrier 2: per-wave results -> thread-0 merge

    // ---- stage 3: thread 0 normalizes + beam update (no trailing barrier:
    // scratch of this parity is not rewritten until step t+2, and every path
    // to those writes crosses barrier 2 of step t+1) --------------------
    if (tid == 0) {
      float bs = red_s[p][0];
#pragma unroll
      for (int w = 1; w < NW_; ++w) bs += red_s[p][w];
      const float inv = 1.0f / bs;

      u64 k0 = wkey[p][0][0], k1 = wkey[p][0][1], k2 = wkey[p][0][2];
#pragma unroll
      for (int w = 1; w < NW_; ++w)
        merge3u(k0, k1, k2, wkey[p][w][0], wkey[p][w][1], wkey[p][w][2]);

      const u64 kk[3] = {k0, k1, k2};
      float lp[3]; int vv[3];
#pragma unroll
      for (int j = 0; j < 3; ++j) {
        float ee = __uint_as_float((unsigned)(kk[j] >> 32));
        vv[j] = (V_ - 1) - (int)(kk[j] & 0xffffffffu);
        lp[j] = __logf(ee * inv + EPS_F);  // log(softmax + eps)
      }

      // exact top-3 over all 3*V candidates == top-3 of this 3x3 set,
      // tie-break on flattened index k*V + v (beam-major, like the ref)
      float f0 = -__builtin_inff(), f1 = f0, f2 = f0;
      int   g0 = 0x7fffffff, g1 = g0, g2 = g0;
      const float ss[3] = {s0, s1, s2};
#pragma unroll
      for (int k = 0; k < 3; ++k)
#pragma unroll
        for (int j = 0; j < 3; ++j)
          finsert3(ss[k] + lp[j], k * V_ + vv[j], f0, g0, f1, g1, f2, g2);

      s0 = f0; s1 = f1; s2 = f2;
      pars[t * K_ + 0] = (unsigned char)(g0 >> 10); toks[t * K_ + 0] = (short)(g0 & (V_ - 1));
      pars[t * K_ + 1] = (unsigned char)(g1 >> 10); toks[t * K_ + 1] = (short)(g1 & (V_ - 1));
      pars[t * K_ + 2] = (unsigned char)(g2 >> 10); toks[t * K_ + 2] = (short)(g2 & (V_ - 1));
    }
  }

  if (tid == 0) { sc_fin[0] = s0; sc_fin[1] = s1; sc_fin[2] = s2; }
  __syncthreads();  // publish pars/toks/sc_fin to the backtrack threads

  // ---- backtrack: reconstruct histories (equivalent to tokens[:,parent]) --
  if (tid < K_) {
    int beam = tid;
    for (int t = T_ - 1; t >= 0; --t) {
      tok_out[((size_t)b * T_ + t) * K_ + tid] = (int)toks[t * K_ + beam];
      beam = pars[t * K_ + beam];
    }
    sc_out[(size_t)b * K_ + tid] = sc_fin[tid];
  }
}

extern "C" void kernel_launch(void* const* d_in, const int* in_sizes, int n_in,
                              void* d_out, int out_size, void* d_ws, size_t ws_size,
                              hipStream_t stream) {
  (void)n_in; (void)out_size; (void)d_ws; (void)ws_size;
  const float* logits = (const float*)d_in[0];
  const int B = in_sizes[0] / (T_ * V_);  // 128
  int*   tok_out = (int*)d_out;                                                 // (B,T,K) i32
  float* sc_out  = (float*)((char*)d_out + (size_t)B * T_ * K_ * sizeof(int));  // (B,K) f32
  beam_kernel<<<dim3(B), dim3(NT_), 0, stream>>>(logits, tok_out, sc_out);
}